// XFeat_38250978738188
// MI455X (gfx1250) — compile-verified
//
#include <hip/hip_runtime.h>
#include <hip/hip_bf16.h>
#include <stdint.h>

#define HC 272
#define WC 480
#define HH (HC*8)    // 2176
#define WH (WC*8)    // 3840
#define NP (HH*WH)   // 8355840 pixels  (divisible by 256)
#define NC (HC*WC)   // 130560 cells
#define TOPK 4096
#define NMS_TH 0.05f
#define CUBIC_A (-0.75f)

typedef __attribute__((ext_vector_type(2))) float    v2f;
typedef __attribute__((ext_vector_type(8))) float    v8f;
typedef __attribute__((ext_vector_type(4))) unsigned v4u;
typedef __attribute__((ext_vector_type(8))) int      v8i;
typedef __attribute__((ext_vector_type(4))) int      v4i;

// ---------------- kernel 1: softmax(65) -> 8x8 pixel shuffle heatmap ----------------
__global__ void __launch_bounds__(256) k_heatmap(const float* __restrict__ K1,
                                                 float* __restrict__ K1h) {
    int cell = blockIdx.x * blockDim.x + threadIdx.x;
    if (cell >= NC) return;
    int yc = cell / WC, xc = cell % WC;
    float v[65];
    float m = -INFINITY;
#pragma unroll
    for (int c = 0; c < 65; ++c) { v[c] = K1[c * NC + cell]; m = fmaxf(m, v[c]); }
    float s = 0.f;
#pragma unroll
    for (int c = 0; c < 65; ++c) { v[c] = __expf(v[c] - m); s += v[c]; }
    float inv = 1.0f / s;
#pragma unroll
    for (int c = 0; c < 64; ++c) {
        int r = c >> 3, cc = c & 7;   // channel -> (row, col) inside 8x8 block
        K1h[(size_t)(yc * 8 + r) * WH + xc * 8 + cc] = v[c] * inv;
    }
}

// ---------------- kernel 2: inverse channel L2-norm of M1 per cell ----------------
__global__ void __launch_bounds__(256) k_invnorm(const float* __restrict__ M1,
                                                 float* __restrict__ invn) {
    int cell = blockIdx.x * blockDim.x + threadIdx.x;
    if (cell >= NC) return;
    float ss = 0.f;
#pragma unroll
    for (int c = 0; c < 64; ++c) { float t = M1[c * NC + cell]; ss = fmaf(t, t, ss); }
    invn[cell] = 1.0f / fmaxf(sqrtf(ss), 1e-12f);
}

// ---------------- kernel 3: 5x5 NMS + score -> keys, fused round-0 histogram ----------------
__device__ __forceinline__ float heat_at(const float* __restrict__ h, int y, int x) {
    if ((unsigned)y >= (unsigned)HH || (unsigned)x >= (unsigned)WH) return 0.f;
    return h[(size_t)y * WH + x];
}

__global__ void __launch_bounds__(256) k_nms_score(const float* __restrict__ K1h,
                                                   const float* __restrict__ H1,
                                                   unsigned* __restrict__ keys,
                                                   unsigned* __restrict__ hist) {
    __shared__ unsigned lh[256];
    lh[threadIdx.x] = 0;
    __syncthreads();

    int i = blockIdx.x * blockDim.x + threadIdx.x;   // NP % 256 == 0: no stragglers
    int x = i % WH, y = i / WH;
    float v = K1h[i];
    float score = -1.0f;
    bool kpt = (v > NMS_TH);
    if (kpt) {
        float m = -INFINITY;
#pragma unroll
        for (int dy = -2; dy <= 2; ++dy) {
#pragma unroll
            for (int dx = -2; dx <= 2; ++dx) {
                int yy = y + dy, xx = x + dx;
                if ((unsigned)yy < (unsigned)HH && (unsigned)xx < (unsigned)WH)
                    m = fmaxf(m, K1h[(size_t)yy * WH + xx]);
            }
        }
        kpt = (v >= m);   // == local max (m >= v always)
    }
    if (kpt && !(x == 0 && y == 0)) {
        // nearest interp of K1h at (x,y): ix = x*W/(W-1) - 0.5, round-half-even
        float ixn = (float)x * ((float)WH / (float)(WH - 1)) - 0.5f;
        float iyn = (float)y * ((float)HH / (float)(HH - 1)) - 0.5f;
        float sc_n = heat_at(K1h, (int)rintf(iyn), (int)rintf(ixn));
        // bilinear interp of H1 (Hc x Wc) with zero OOB taps
        float ixb = (float)x * ((float)WC / (float)(WH - 1)) - 0.5f;
        float iyb = (float)y * ((float)HC / (float)(HH - 1)) - 0.5f;
        float x0f = floorf(ixb), y0f = floorf(iyb);
        float tx = ixb - x0f, ty = iyb - y0f;
        int x0 = (int)x0f, y0 = (int)y0f;
        auto g = [&](int yy, int xx) -> float {
            if ((unsigned)yy >= (unsigned)HC || (unsigned)xx >= (unsigned)WC) return 0.f;
            return H1[yy * WC + xx];
        };
        float sc_b = g(y0, x0) * (1.f - tx) * (1.f - ty) + g(y0, x0 + 1) * tx * (1.f - ty)
                   + g(y0 + 1, x0) * (1.f - tx) * ty + g(y0 + 1, x0 + 1) * tx * ty;
        score = sc_n * sc_b;
    }
    unsigned u = __float_as_uint(score);
    u = (u & 0x80000000u) ? ~u : (u | 0x80000000u);   // monotone map, larger f -> larger u
    keys[i] = u;
    atomicAdd(&lh[u >> 24], 1u);                      // fused radix round-0 histogram

    __syncthreads();
    if (lh[threadIdx.x]) atomicAdd(&hist[threadIdx.x], lh[threadIdx.x]);
}

// ---------------- top-k: exact 4-round radix select ----------------
__global__ void k_init(unsigned* __restrict__ hist, unsigned* __restrict__ state) {
    int t = threadIdx.x;
    if (t < 256) hist[t] = 0;
    if (t == 0) { state[0] = 0; state[1] = TOPK; state[2] = 0; state[3] = 0; }
}

__global__ void __launch_bounds__(256) k_hist(const unsigned* __restrict__ keys,
                                              const unsigned* __restrict__ state,
                                              unsigned* __restrict__ hist, int round) {
    __shared__ unsigned lh[256];
    lh[threadIdx.x] = 0;
    __syncthreads();
    unsigned prefix = state[0];
    int shift = 24 - 8 * round;
    unsigned stride = gridDim.x * blockDim.x;
    for (unsigned i = blockIdx.x * blockDim.x + threadIdx.x; i < NP; i += stride) {
        __builtin_prefetch(keys + i + stride, 0, 0);      // global_prefetch_b8
        unsigned k = keys[i];
        if ((k >> (shift + 8)) == (prefix >> (shift + 8)))
            atomicAdd(&lh[(k >> shift) & 0xFFu], 1u);
    }
    __syncthreads();
    if (lh[threadIdx.x]) atomicAdd(&hist[threadIdx.x], lh[threadIdx.x]);
}

__global__ void k_scan(unsigned* __restrict__ hist, unsigned* __restrict__ state, int round) {
    __shared__ unsigned sh[256];
    sh[threadIdx.x] = hist[threadIdx.x];
    __syncthreads();
    if (threadIdx.x == 0) {
        unsigned remaining = state[1];
        unsigned acc = 0; int b = 0;
        for (int i = 255; i >= 0; --i) {
            if (acc + sh[i] >= remaining) { b = i; break; }
            acc += sh[i];
        }
        int shift = 24 - 8 * round;
        state[0] |= ((unsigned)b) << shift;
        state[1] = remaining - acc;   // elements still needed from this bin downward
    }
    __syncthreads();
    hist[threadIdx.x] = 0;            // ready for next round
}

__global__ void __launch_bounds__(256) k_compact(const unsigned* __restrict__ keys,
                                                 unsigned* __restrict__ state,
                                                 unsigned long long* __restrict__ sortbuf) {
    unsigned kth = state[0];          // exact value of the K-th largest key
    unsigned remaining = state[1];    // how many ==kth entries to take
    unsigned stride = gridDim.x * blockDim.x;
    for (unsigned i = blockIdx.x * blockDim.x + threadIdx.x; i < NP; i += stride) {
        unsigned k = keys[i];
        if (k > kth) {
            unsigned p = atomicAdd(&state[2], 1u);           // exactly TOPK-remaining of these
            if (p < TOPK) sortbuf[p] = ((unsigned long long)k << 32) | (unsigned)(~i);
        } else if (k == kth) {
            unsigned e = atomicAdd(&state[3], 1u);
            if (e < remaining)
                sortbuf[(TOPK - remaining) + e] = ((unsigned long long)k << 32) | (unsigned)(~i);
        }
    }
}

// bitonic sort of 4096 packed (key desc, idx asc) entries in LDS (32 KB)
__global__ void __launch_bounds__(1024) k_sort(unsigned long long* __restrict__ sortbuf) {
    __shared__ unsigned long long s[TOPK];
    for (int i = threadIdx.x; i < TOPK; i += blockDim.x) s[i] = sortbuf[i];
    __syncthreads();
    for (int k = 2; k <= TOPK; k <<= 1) {
        for (int j = k >> 1; j > 0; j >>= 1) {
            for (int i = threadIdx.x; i < TOPK; i += blockDim.x) {
                int ixj = i ^ j;
                if (ixj > i) {
                    bool desc = ((i & k) == 0);
                    unsigned long long a = s[i], b = s[ixj];
                    if (desc ? (a < b) : (a > b)) { s[i] = b; s[ixj] = a; }
                }
            }
            __syncthreads();
        }
    }
    for (int i = threadIdx.x; i < TOPK; i += blockDim.x) sortbuf[i] = s[i];
}

// ---------------- outputs: mkpts, scores, valid ----------------
__global__ void __launch_bounds__(256) k_write(const unsigned long long* __restrict__ sortbuf,
                                               float* __restrict__ out) {
    int t = blockIdx.x * blockDim.x + threadIdx.x;
    if (t >= TOPK) return;
    unsigned long long v = sortbuf[t];
    unsigned idx = ~(unsigned)(v & 0xFFFFFFFFull);
    unsigned key = (unsigned)(v >> 32);
    unsigned orig = (key & 0x80000000u) ? (key ^ 0x80000000u) : ~key;
    float score = __uint_as_float(orig);
    int x = (int)(idx % WH), y = (int)(idx / WH);
    bool kpt = (score >= 0.f);                       // non-kpts carry exactly -1
    out[2 * t]     = kpt ? (float)x : 0.f;           // mkpts_k
    out[2 * t + 1] = kpt ? (float)y : 0.f;
    out[2 * TOPK + t] = score;                       // top_scores
    out[2 * TOPK + TOPK + TOPK * 64 + t] = (score > 0.f) ? 1.f : 0.f;  // valid
}

// ---------------- bicubic gather of L2-normalized descriptors ----------------
__device__ __forceinline__ float cubic_w(float t) {
    float at = fabsf(t);
    float w1 = ((CUBIC_A + 2.f) * at - (CUBIC_A + 3.f)) * at * at + 1.f;
    float w2 = CUBIC_A * (((at - 5.f) * at + 8.f) * at - 4.f);
    return (at <= 1.f) ? w1 : ((at < 2.f) ? w2 : 0.f);
}

__global__ void __launch_bounds__(64) k_feats(const float* __restrict__ M1,
                                              const float* __restrict__ invn,
                                              const float* __restrict__ out,
                                              float* __restrict__ feats) {
    int p = blockIdx.x;       // keypoint
    int c = threadIdx.x;      // channel
    float px = out[2 * p], py = out[2 * p + 1];
    float ix = px * ((float)WC / (float)(WH - 1)) - 0.5f;
    float iy = py * ((float)HC / (float)(HH - 1)) - 0.5f;
    float x0f = floorf(ix), y0f = floorf(iy);
    float tx = ix - x0f, ty = iy - y0f;
    int x0 = (int)x0f, y0 = (int)y0f;
    float acc = 0.f;
#pragma unroll
    for (int j = -1; j <= 2; ++j) {
        float wy = cubic_w(ty - (float)j);
        int yy = y0 + j;
#pragma unroll
        for (int i2 = -1; i2 <= 2; ++i2) {
            float wx = cubic_w(tx - (float)i2);
            int xx = x0 + i2;
            float val = 0.f;
            if ((unsigned)yy < (unsigned)HC && (unsigned)xx < (unsigned)WC) {
                int cell = yy * WC + xx;
                val = M1[c * NC + cell] * invn[cell];
            }
            acc = fmaf(wy * wx, val, acc);
        }
    }
    feats[p * 64 + c] = acc;
}

// ---------------- TDM: DMA a 2D f32 tile (tile_d0 x tile_d1) from global into LDS ----------------
__device__ __forceinline__ void tdm_load_tile_2d(unsigned lds_off, const void* gptr,
                                                 unsigned tensor_d0, unsigned tensor_d1,
                                                 unsigned stride0,
                                                 unsigned tile_d0, unsigned tile_d1) {
#if __has_builtin(__builtin_amdgcn_tensor_load_to_lds)
    unsigned long long ga = (unsigned long long)(uintptr_t)gptr;
    v4u g0;
    g0[0] = 1u;                                            // count=1, user mode, no gather
    g0[1] = lds_off;                                       // lds_addr (bytes)
    g0[2] = (unsigned)(ga & 0xFFFFFFFFu);                  // global_addr[31:0]
    g0[3] = (unsigned)((ga >> 32) & 0x01FFFFFFu)           // global_addr[56:32]
          | (2u << 30);                                    // type = 2 ("image")
    v8i g1;
    g1[0] = (int)(2u << 16);                               // wg_mask=0, data_size=2 (4 bytes)
    g1[1] = (int)((tensor_d0 & 0xFFFFu) << 16);            // tensor_dim0[15:0] @ bits 63:48
    g1[2] = (int)(((tensor_d0 >> 16) & 0xFFFFu)            // tensor_dim0[31:16]
          |       ((tensor_d1 & 0xFFFFu) << 16));          // tensor_dim1[15:0]
    g1[3] = (int)(((tensor_d1 >> 16) & 0xFFFFu)            // tensor_dim1[31:16]
          |       ((tile_d0 & 0xFFFFu) << 16));            // tile_dim0
    g1[4] = (int)(tile_d1 & 0xFFFFu);                      // tile_dim1 (tile_dim2 = 0)
    g1[5] = (int)stride0;                                  // tensor_dim0_stride[31:0]
    g1[6] = 0;                                             // stride0 hi / dim1_stride lo
    g1[7] = 0;
    v4i z4 = {0, 0, 0, 0};
    v8i z8 = {0, 0, 0, 0, 0, 0, 0, 0};
    // clang-23 / therock-10.0 6-arg form: (g0, g1, g2, g3, g4, cpol)
    __builtin_amdgcn_tensor_load_to_lds(g0, g1, z4, z4, z8, 0);
    __builtin_amdgcn_s_wait_tensorcnt(0);
#else
    // Fallback: cooperative plain load (one wave)
    const float* gp = (const float*)gptr;
    float* dst = (float*)((char*)nullptr + lds_off);       // unused path
    for (unsigned r = 0; r < tile_d1; ++r)
        for (unsigned cI = threadIdx.x; cI < tile_d0; cI += 32)
            dst[r * tile_d0 + cI] = gp[(size_t)r * stride0 + cI];
    __builtin_amdgcn_s_wait_tensorcnt(0);
#endif
}

// ---------------- row L2-normalize via TDM tile staging + WMMA Gram diagonal ----------------
// One wave per 16 feature rows. TDM DMAs the 16x64 f32 tile into LDS; then
// G = F·F^T accumulated with 16 chained V_WMMA_F32_16X16X4_F32 (K=64 in steps
// of 4); diag(G)[m] = ||F_m||^2.
__global__ void __launch_bounds__(32) k_featnorm_wmma(const float* __restrict__ feats,
                                                      float* __restrict__ out) {
    __shared__ float Ftile[16 * 64];
    __shared__ float g[32][8];

    int tile = blockIdx.x;
    int lane = threadIdx.x;
    const float* Fg = feats + (size_t)tile * 16 * 64;

    // async-tensor DMA: 16 rows x 64 cols of f32, row stride 64, into LDS
    tdm_load_tile_2d((unsigned)(uintptr_t)&Ftile[0], Fg,
                     /*tensor_d0=*/64, /*tensor_d1=*/TOPK,
                     /*stride0=*/64, /*tile_d0=*/64, /*tile_d1=*/16);
    __syncthreads();

    int m = lane & 15;
    int khalf = (lane >> 4) * 2;     // A layout: VGPR0 = K={0,2}, VGPR1 = K={1,3}
    v8f cacc = {};
#pragma unroll
    for (int s = 0; s < 16; ++s) {
        v2f a;
        a.x = Ftile[m * 64 + s * 4 + khalf];
        a.y = Ftile[m * 64 + s * 4 + khalf + 1];
        // Gram is symmetric; A-tile doubles as B-tile (same K-split layout).
        cacc = __builtin_amdgcn_wmma_f32_16x16x4_f32(false, a, false, a,
                                                     (short)0, cacc, false, false);
    }
#pragma unroll
    for (int v = 0; v < 8; ++v) g[lane][v] = cacc[v];
    __syncthreads();
    if (lane < 16) {
        // C/D layout: rows 0-7 diag on lanes 0-7 (vgpr=row);
        //             rows 8-15 diag on lanes 24-31 (vgpr=row-8).
        float diag = (lane < 8) ? g[lane][lane] : g[lane + 16][lane - 8];
        float inv = 1.0f / fmaxf(sqrtf(diag), 1e-12f);
        float* dst = out + 2 * TOPK + TOPK + (size_t)(tile * 16 + lane) * 64;
        for (int cidx = 0; cidx < 64; ++cidx)
            dst[cidx] = Ftile[lane * 64 + cidx] * inv;
    }
}

// ---------------- launch ----------------
extern "C" void kernel_launch(void* const* d_in, const int* in_sizes, int n_in,
                              void* d_out, int out_size, void* d_ws, size_t ws_size,
                              hipStream_t stream) {
    const float* M1 = (const float*)d_in[0];   // (1,64,272,480)
    const float* K1 = (const float*)d_in[1];   // (1,65,272,480)
    const float* H1 = (const float*)d_in[2];   // (1,1,272,480)
    float* out = (float*)d_out;                // mkpts(8192) | scores(4096) | feats(262144) | valid(4096)

    char* ws = (char*)d_ws;
    float*              K1h     = (float*)(ws);
    unsigned*           keys    = (unsigned*)(ws + (size_t)NP * 4);
    float*              invn    = (float*)(ws + (size_t)NP * 8);
    float*              feats   = (float*)(ws + (size_t)NP * 8 + (size_t)NC * 4);
    unsigned long long* sortbuf = (unsigned long long*)(ws + (size_t)NP * 8 + (size_t)NC * 4
                                                           + (size_t)TOPK * 64 * 4);
    unsigned*           hist    = (unsigned*)((char*)sortbuf + (size_t)TOPK * 8);
    unsigned*           state   = hist + 256;  // [prefix, remaining, cntGT, cntEQ]

    k_heatmap  <<<(NC + 255) / 256, 256, 0, stream>>>(K1, K1h);
    k_invnorm  <<<(NC + 255) / 256, 256, 0, stream>>>(M1, invn);

    k_init<<<1, 256, 0, stream>>>(hist, state);            // zero hist before fused round 0
    k_nms_score<<<NP / 256, 256, 0, stream>>>(K1h, H1, keys, hist);
    k_scan<<<1, 256, 0, stream>>>(hist, state, 0);         // consume fused round-0 histogram
    for (int r = 1; r < 4; ++r) {
        k_hist<<<2048, 256, 0, stream>>>(keys, state, hist, r);
        k_scan<<<1, 256, 0, stream>>>(hist, state, r);
    }
    k_compact<<<2048, 256, 0, stream>>>(keys, state, sortbuf);
    k_sort   <<<1, 1024, 0, stream>>>(sortbuf);
    k_write  <<<(TOPK + 255) / 256, 256, 0, stream>>>(sortbuf, out);

    k_feats        <<<TOPK, 64, 0, stream>>>(M1, invn, out, feats);
    k_featnorm_wmma<<<TOPK / 16, 32, 0, stream>>>(feats, out);
}